// Embedding_38319698214986
// MI455X (gfx1250) — compile-verified
//
#include <hip/hip_runtime.h>
#include <cstdint>
#include <cstddef>

// ---------------------------------------------------------------------------
// Embedding gather: out[r, :] = weight[idx[r], :]
//   idx    : int32  [16384]          (values < 50257 < 2^16 -> 16-bit TDM idx)
//   weight : float32[50257, 1024]
//   out    : float32[16384, 1024]
// Zero-FLOP, HBM-bound: 64 MiB gathered read + 64 MiB write ~= 5.8 us @23.3TB/s.
// Main path: CDNA5 TDM gather-mode (tensor_load_to_lds) - 16 rows / descriptor.
// ---------------------------------------------------------------------------

#define FEATURES      1024
#define PAGES_CONST   50257u
#define ROWS_PER_TDM  16

typedef __attribute__((ext_vector_type(4))) unsigned int v4u;
typedef __attribute__((ext_vector_type(4))) int          v4i;
typedef __attribute__((ext_vector_type(8))) int          v8i;
typedef __attribute__((ext_vector_type(4))) float        f4;

#if __has_builtin(__builtin_amdgcn_tensor_load_to_lds)
#define HAVE_TDM 1
#else
#define HAVE_TDM 0
#endif

#if HAVE_TDM
__global__ __launch_bounds__(256) void embed_gather_tdm(
    const int* __restrict__ idx, const float* __restrict__ w,
    float* __restrict__ out)
{
  // 16 rows x 1024 f32 = 64 KiB staging tile (only LDS object -> offset 0).
  __shared__ __align__(16) float tile[ROWS_PER_TDM * FEATURES];

  const unsigned b = blockIdx.x;

  if (threadIdx.x < 32) {  // wave 0 drives the Tensor Data Mover
    const int* ip = idx + (size_t)b * ROWS_PER_TDM;
    unsigned i0  = (unsigned)ip[0],  i1  = (unsigned)ip[1];
    unsigned i2  = (unsigned)ip[2],  i3  = (unsigned)ip[3];
    unsigned i4  = (unsigned)ip[4],  i5  = (unsigned)ip[5];
    unsigned i6  = (unsigned)ip[6],  i7  = (unsigned)ip[7];
    unsigned i8  = (unsigned)ip[8],  i9  = (unsigned)ip[9];
    unsigned i10 = (unsigned)ip[10], i11 = (unsigned)ip[11];
    unsigned i12 = (unsigned)ip[12], i13 = (unsigned)ip[13];
    unsigned i14 = (unsigned)ip[14], i15 = (unsigned)ip[15];

    unsigned long long ga = (unsigned long long)(const void*)w;

    // ---- D# group 0 (ISA 08 §8.3) ----
    // w0: count=1 | gather_index_size=0 (16-bit) <<30 | gather_mode=1 <<31
    // w1: lds_addr = 0 (tile is at LDS offset 0)
    // w2: global_addr[31:0]; w3: global_addr[56:32] | type=2 <<30
    v4u g0 = {
      0x80000001u,
      0u,
      (unsigned)ga,
      (unsigned)((ga >> 32) & 0x01FFFFFFull) | 0x80000000u
    };

    // ---- D# group 1 (ISA 08 §8.4) ----
    // w0: workgroup_mask=0 | data_size=2 (4 B) <<16
    // w1: atomic_barrier_addr=0 | tensor_dim0[15:0]<<16         (1024)
    // w2: tensor_dim0[31:16] | tensor_dim1[15:0]<<16            (50257)
    // w3: tensor_dim1[31:16] | tile_dim0<<16                    (1024)
    // w4: tile_dim1 = 16 (#valid gather indices); tile_dim2 ignored
    // w5: tensor_dim0_stride[31:0] = 1024; w6: stride[47:32]=0 | dim1_stride ign.
    v8i g1 = {
      (int)0x00020000u,
      (int)((FEATURES & 0xFFFFu) << 16),
      (int)(((PAGES_CONST & 0xFFFFu) << 16) | ((FEATURES >> 16) & 0xFFFFu)),
      (int)(((FEATURES & 0xFFFFu) << 16) | ((PAGES_CONST >> 16) & 0xFFFFu)),
      (int)ROWS_PER_TDM,
      (int)FEATURES,
      0,
      0
    };

    // ---- D# groups 2/3: sixteen 16-bit row indices (§8.7) ----
    v4i g2 = {
      (int)((i0  & 0xFFFFu) | (i1  << 16)),
      (int)((i2  & 0xFFFFu) | (i3  << 16)),
      (int)((i4  & 0xFFFFu) | (i5  << 16)),
      (int)((i6  & 0xFFFFu) | (i7  << 16))
    };
    v4i g3 = {
      (int)((i8  & 0xFFFFu) | (i9  << 16)),
      (int)((i10 & 0xFFFFu) | (i11 << 16)),
      (int)((i12 & 0xFFFFu) | (i13 << 16)),
      (int)((i14 & 0xFFFFu) | (i15 << 16))
    };

    // 6-arg form on this toolchain (clang-23 / therock-10.0 headers):
    // (g0, g1, g2, g3, <uncharacterized v8i — ISA defines no 5th D# group,
    //  VADDR4 is documented unused -> zero-fill>, cpol)
    v8i gz = { 0, 0, 0, 0, 0, 0, 0, 0 };
    __builtin_amdgcn_tensor_load_to_lds(g0, g1, g2, g3, gz, /*cpol=*/0);

#if __has_builtin(__builtin_amdgcn_s_wait_tensorcnt)
    __builtin_amdgcn_s_wait_tensorcnt(0);
#else
    asm volatile("s_wait_tensorcnt 0x0" ::: "memory");
#endif
  }
  __syncthreads();  // publish LDS tile to all 8 waves

  // The TDM builtin is not modeled as writing LDS. A bare "memory" clobber is
  // NOT enough: `tile` is an internal non-escaping LDS global, so AA still
  // proves it store-free and folds the loads to undef (observed round 2/3).
  // Passing the pointer as an asm operand ESCAPES it -> the asm may write it
  // -> the ds_loads below must be materialized.
  asm volatile("" :: "s"(tile) : "memory");

  // Stream LDS -> out: 4096 f4 per block, 16 per thread.
  // Non-temporal stores: output is write-once, keep L2 for the weight table.
  const f4* sv = (const f4*)tile;
  f4* ov = (f4*)(out + (size_t)b * (ROWS_PER_TDM * FEATURES));
  const int t = (int)threadIdx.x;
#pragma unroll
  for (int j = 0; j < (ROWS_PER_TDM * FEATURES / 4) / 256; ++j) {
    int e = j * 256 + t;
    __builtin_nontemporal_store(sv[e], ov + e);
  }
}
#else
// Fallback: direct vectorized gather (same roofline: one b128 load + one
// non-temporal b128 store per 16 output bytes).
__global__ __launch_bounds__(256) void embed_gather_direct(
    const int* __restrict__ idx, const float* __restrict__ w,
    float* __restrict__ out, int nrows)
{
  int e = blockIdx.x * 256 + (int)threadIdx.x;      // f4 element index
  int total = nrows * (FEATURES / 4);
  if (e >= total) return;
  int row = e >> 8;            // FEATURES/4 == 256 f4 per row
  int col = e & 255;
  int r = idx[row];
  const f4* src = (const f4*)w + (size_t)r * (FEATURES / 4) + col;
  __builtin_nontemporal_store(*src, (f4*)out + e);
}
#endif

extern "C" void kernel_launch(void* const* d_in, const int* in_sizes, int n_in,
                              void* d_out, int out_size, void* d_ws, size_t ws_size,
                              hipStream_t stream) {
  const int*   idx = (const int*)d_in[0];    // [8*2048] int32
  const float* w   = (const float*)d_in[1];  // [50257*1024] f32
  float*       out = (float*)d_out;          // [8*2048*1024] f32
  const int nrows  = in_sizes[0];            // 16384

#if HAVE_TDM
  const int groups = nrows / ROWS_PER_TDM;   // 1024 blocks, 16 rows each
  embed_gather_tdm<<<groups, 256, 0, stream>>>(idx, w, out);
#else
  const int total4 = nrows * (FEATURES / 4);
  embed_gather_direct<<<(total4 + 255) / 256, 256, 0, stream>>>(idx, w, out, nrows);
#endif
}